// SigmoidDVHLoss_66949950210775
// MI455X (gfx1250) — compile-verified
//
#include <hip/hip_runtime.h>
#include <hip/hip_bf16.h>

typedef float v2f __attribute__((ext_vector_type(2)));
typedef float v8f __attribute__((ext_vector_type(8)));

#define NPTV 3
#define NBATCH 8
#define NACC 9            // sigP0..2, sigT0..2, cnt0..2
#define BLK 256
#define GRIDX 256
#define LDS_STRIDE 17     // pad 16 -> 17 dwords to avoid 64-bank conflicts

__device__ __forceinline__ float fast_sigmoid(float z) {
    // sigmoid(z) = 1 / (1 + exp(-z)) ; exp via native v_exp_f32 (TRANS, co-executes)
    float e = __builtin_amdgcn_exp2f(-z * 1.4426950408889634f);
    return __builtin_amdgcn_rcpf(1.0f + e);
}

__global__ void dvh_ws_zero(float* ws) {
    if (threadIdx.x < NBATCH * NACC) ws[threadIdx.x] = 0.0f;
}

__global__ void __launch_bounds__(BLK)
dvh_partial(const float* __restrict__ pred,
            const float* __restrict__ targ,
            const int*   __restrict__ mask,
            const float* __restrict__ ptv,
            float* __restrict__ ws,
            int v4_per_batch) {
    __shared__ float red[BLK * LDS_STRIDE];

    const int tid   = threadIdx.x;
    const int batch = blockIdx.y;

    // uniform scalar loads of the 3 thresholds
    const float p0 = ptv[0], p1 = ptv[1], p2 = ptv[2];

    const float4* __restrict__ pr = reinterpret_cast<const float4*>(pred) + (size_t)batch * v4_per_batch;
    const float4* __restrict__ tg = reinterpret_cast<const float4*>(targ) + (size_t)batch * v4_per_batch;
    const int4*   __restrict__ mk = reinterpret_cast<const int4*>(mask)   + (size_t)batch * v4_per_batch;

    float aP0 = 0.f, aP1 = 0.f, aP2 = 0.f;
    float aT0 = 0.f, aT1 = 0.f, aT2 = 0.f;
    float c0  = 0.f, c1  = 0.f, c2  = 0.f;

    const int stride = GRIDX * BLK;
    for (int j = blockIdx.x * BLK + tid; j < v4_per_batch; j += stride) {
        float4 p = pr[j];              // global_load_b128
        float4 t = tg[j];              // global_load_b128
        int4   m = mk[j];              // global_load_b128

        const float pd[4] = {p.x, p.y, p.z, p.w};
        const float td[4] = {t.x, t.y, t.z, t.w};
        const int   md[4] = {m.x, m.y, m.z, m.w};

#pragma unroll
        for (int e = 0; e < 4; ++e) {
            int   lab = md[e];
            float thr = (lab == 1) ? p0 : ((lab == 2) ? p1 : p2);
            float sp  = fast_sigmoid((pd[e] - thr) * 10.0f);
            float st  = fast_sigmoid((td[e] - thr) * 10.0f);
            bool i1 = (lab == 1), i2 = (lab == 2), i3 = (lab == 3);
            aP0 += i1 ? sp : 0.f;  aP1 += i2 ? sp : 0.f;  aP2 += i3 ? sp : 0.f;
            aT0 += i1 ? st : 0.f;  aT1 += i2 ? st : 0.f;  aT2 += i3 ? st : 0.f;
            c0  += i1 ? 1.f : 0.f; c1  += i2 ? 1.f : 0.f; c2  += i3 ? 1.f : 0.f;
        }
    }

    // dump 9 accumulators per thread to LDS (padded rows)
    float* row = &red[tid * LDS_STRIDE];
    row[0] = aP0; row[1] = aP1; row[2] = aP2;
    row[3] = aT0; row[4] = aT1; row[5] = aT2;
    row[6] = c0;  row[7] = c1;  row[8] = c2;
    __syncthreads();

    // ---- wave 0 only: one V_WMMA_F32_16X16X4_F32 reduces all 9 (of 16 rows)
    // accumulators across the whole block. A(16x4): row m = accumulator m,
    // K-groups = 4 chunks of 64 threads (pre-summed). B = ones => D[m][*] =
    // row sum = block total of accumulator m. Rows 9..15 carry garbage but
    // WMMA rows are independent and we never read them.
    if (tid < 32) {                      // wave32: EXEC all ones inside
        const int col  = tid & 15;       // accumulator index (A row)
        const int tb   = (tid >> 4) * 128;
        float s0 = 0.f, s1 = 0.f;
#pragma unroll 4
        for (int t = 0; t < 64; ++t) {
            s0 += red[(tb + t) * LDS_STRIDE + col];
            s1 += red[(tb + 64 + t) * LDS_STRIDE + col];
        }
        v2f a; a[0] = s0; a[1] = s1;     // lanes 0-15: K=0,1 ; lanes 16-31: K=2,3
        v2f b; b[0] = 1.0f; b[1] = 1.0f; // all-ones B (4x16)
        v8f cacc = {};
        v8f d = __builtin_amdgcn_wmma_f32_16x16x4_f32(
            false, a, false, b, (short)0, cacc, false, false);

        // D layout: VGPR r, lanes 0-15 -> row r ; lanes 16-31 -> row r+8
        float* wsB = ws + batch * NACC;
        if (tid == 0) {
#pragma unroll
            for (int r = 0; r < 8; ++r) atomicAdd(&wsB[r], d[r]);
        } else if (tid == 16) {
            atomicAdd(&wsB[8], d[0]);    // row 8
        }
    }
}

__global__ void dvh_finalize(const float* __restrict__ ws, float* __restrict__ out) {
    if (threadIdx.x == 0) {
        float acc = 0.f;
        for (int b = 0; b < NBATCH; ++b) {
            const float* w = ws + b * NACC;
            for (int k = 0; k < NPTV; ++k) {
                float diff = (w[k] - w[3 + k]) / w[6 + k];  // (sumP - sumT)/cnt
                acc += diff * diff;
            }
        }
        out[0] = acc / (float)(NBATCH * NPTV);
    }
}

extern "C" void kernel_launch(void* const* d_in, const int* in_sizes, int n_in,
                              void* d_out, int out_size, void* d_ws, size_t ws_size,
                              hipStream_t stream) {
    const float* pred = (const float*)d_in[0];
    const float* targ = (const float*)d_in[1];
    const int*   mask = (const int*)d_in[2];
    const float* ptv  = (const float*)d_in[3];
    float* ws  = (float*)d_ws;
    float* out = (float*)d_out;

    const int V  = in_sizes[0] / NBATCH;   // voxels per batch (128^3)
    const int V4 = V / 4;                  // float4 elements per batch

    dvh_ws_zero<<<1, 128, 0, stream>>>(ws);
    dim3 grid(GRIDX, NBATCH);
    dvh_partial<<<grid, BLK, 0, stream>>>(pred, targ, mask, ptv, ws, V4);
    dvh_finalize<<<1, 32, 0, stream>>>(ws, out);
}